// OSM_CAA_Loss_18726057410586
// MI455X (gfx1250) — compile-verified
//
#include <hip/hip_runtime.h>
#include <hip/hip_bf16.h>
#include <math.h>

#define ALPHA  1.2f
#define L_BAL  0.5f
#define SIGMA2 0.64f      // SIGMA^2, SIGMA = 0.8
#define EPSV   1e-12f

typedef __attribute__((ext_vector_type(2))) float v2f;
typedef __attribute__((ext_vector_type(8))) float v8f;

// ---------------- K0: zero the 4 loss accumulators ----------------
__global__ void k_zero_accum(float* accum) {
    if (threadIdx.x < 4) accum[threadIdx.x] = 0.0f;
}

// ---------------- K1: L2-normalize rows of x; emit xn and sq ----------------
__global__ void k_rownorm(const float* __restrict__ x, float* __restrict__ xn,
                          float* __restrict__ sq, int d) {
    const int row = blockIdx.x;
    const float* xr = x + (size_t)row * d;
    float*       xo = xn + (size_t)row * d;
    float s = 0.0f;
    for (int k = threadIdx.x; k < d; k += blockDim.x) { float v = xr[k]; s += v * v; }
    __shared__ float red[256];
    red[threadIdx.x] = s; __syncthreads();
    for (int off = 128; off > 0; off >>= 1) {
        if (threadIdx.x < off) red[threadIdx.x] += red[threadIdx.x + off];
        __syncthreads();
    }
    const float sumsq = red[0];
    const float inv   = 1.0f / fmaxf(sqrtf(sumsq), EPSV);
    for (int k = threadIdx.x; k < d; k += blockDim.x) xo[k] = xr[k] * inv;
    if (threadIdx.x == 0) sq[row] = sumsq * inv * inv;   // == sum(xn^2)
}

// ---------------- K2: L2-normalize columns of embd; write transposed enT ----------------
__global__ void k_colnorm_T(const float* __restrict__ embd, float* __restrict__ enT,
                            int d, int C) {
    const int c = blockIdx.x;
    float s = 0.0f;
    for (int k = threadIdx.x; k < d; k += blockDim.x) {
        float v = embd[(size_t)k * C + c]; s += v * v;
    }
    __shared__ float red[256];
    red[threadIdx.x] = s; __syncthreads();
    for (int off = 128; off > 0; off >>= 1) {
        if (threadIdx.x < off) red[threadIdx.x] += red[threadIdx.x + off];
        __syncthreads();
    }
    const float inv = 1.0f / fmaxf(sqrtf(red[0]), EPSV);
    for (int k = threadIdx.x; k < d; k += blockDim.x)
        enT[(size_t)c * d + k] = embd[(size_t)k * C + c] * inv;
}

// ---------------- K3: logits + softmax + atten gather (one row per block) ----------------
__global__ void k_atten(const float* __restrict__ xn, const float* __restrict__ enT,
                        const long long* __restrict__ labels, float* __restrict__ atten,
                        int d, int C) {
    __shared__ float xrow[2048];
    __shared__ float logits[1000];
    __shared__ float red[256];
    const int i   = blockIdx.x;
    const int tid = threadIdx.x;
    for (int k = tid; k < d; k += blockDim.x) xrow[k] = xn[(size_t)i * d + k];
    __syncthreads();
    for (int c = tid; c < C; c += blockDim.x) {
        const float* w = enT + (size_t)c * d;
        float acc = 0.0f;
        #pragma unroll 8
        for (int k = 0; k < d; ++k) acc = fmaf(xrow[k], w[k], acc);
        logits[c] = acc;
    }
    __syncthreads();
    float mx = -3.4e38f;
    for (int c = tid; c < C; c += blockDim.x) mx = fmaxf(mx, logits[c]);
    red[tid] = mx; __syncthreads();
    for (int off = 128; off > 0; off >>= 1) {
        if (tid < off) red[tid] = fmaxf(red[tid], red[tid + off]);
        __syncthreads();
    }
    const float M = red[0];
    __syncthreads();
    float se = 0.0f;
    for (int c = tid; c < C; c += blockDim.x) se += expf(logits[c] - M);
    red[tid] = se; __syncthreads();
    for (int off = 128; off > 0; off >>= 1) {
        if (tid < off) red[tid] += red[tid + off];
        __syncthreads();
    }
    if (tid == 0) {
        const int lab = (int)labels[i];
        atten[i] = expf(logits[lab] - M) / red[0];
    }
}

// ---------------- K4: fp32 WMMA Gram (64x64/block, 2x2 reg-blocked) + fused loss ----------------
#define BT   64    // block tile (rows == cols)
#define KBLK 64
#define LDK  68    // padded stride (floats): 272B rows -> 16B aligned, conflict-free

__global__ void k_gram_loss(const float* __restrict__ xn, const float* __restrict__ sq,
                            const float* __restrict__ atten,
                            const long long* __restrict__ labels,
                            float* __restrict__ accum, int n, int d) {
    __shared__ float As[BT * LDK];
    __shared__ float Bs[BT * LDK];
    __shared__ float attI[BT], attJ[BT], sqI[BT], sqJ[BT];
    __shared__ int   labI[BT], labJ[BT];
    __shared__ float red[4 * 128];

    const int i0  = blockIdx.x * BT;
    const int j0  = blockIdx.y * BT;
    const int tid = threadIdx.x;

    if (tid < BT) {
        attI[tid] = atten[i0 + tid]; sqI[tid] = sq[i0 + tid]; labI[tid] = (int)labels[i0 + tid];
    } else {
        const int t = tid - BT;
        attJ[t] = atten[j0 + t]; sqJ[t] = sq[j0 + t]; labJ[t] = (int)labels[j0 + t];
    }

    const int wave = tid >> 5;
    const int lane = tid & 31;
    const int half = lane >> 4;          // K pair {0,1} vs {2,3}
    const int ln   = lane & 15;          // M (A-frag) / N (B-frag) index
    const int wr   = (wave >> 1) * 32;   // wave row offset within block tile
    const int wc   = (wave & 1) * 32;    // wave col offset within block tile

    v8f acc00 = {}, acc01 = {}, acc10 = {}, acc11 = {};

    const int row0 = (wr + ln) * LDK;        // A subtile 0 row base
    const int row1 = (wr + 16 + ln) * LDK;   // A subtile 1 row base
    const int col0 = (wc + ln) * LDK;        // B subtile 0 row base
    const int col1 = (wc + 16 + ln) * LDK;   // B subtile 1 row base

    for (int k0 = 0; k0 < d; k0 += KBLK) {
        __syncthreads();
        // 64x64 tile = 1024 float4s; 128 threads x 8 float4 each, coalesced
        for (int idx = tid; idx < BT * (KBLK / 4); idx += 128) {
            const int r  = idx >> 4;
            const int c4 = (idx & 15) * 4;
            const float4 av = *(const float4*)(xn + (size_t)(i0 + r) * d + k0 + c4);
            const float4 bv = *(const float4*)(xn + (size_t)(j0 + r) * d + k0 + c4);
            *(float4*)(As + r * LDK + c4) = av;
            *(float4*)(Bs + r * LDK + c4) = bv;
        }
        __syncthreads();
        #pragma unroll
        for (int kk = 0; kk < KBLK; kk += 4) {
            const int ka = kk + 2 * half;
            v2f a0, a1, b0, b1;
            a0.x = As[row0 + ka]; a0.y = As[row0 + ka + 1];
            a1.x = As[row1 + ka]; a1.y = As[row1 + ka + 1];
            b0.x = Bs[col0 + ka]; b0.y = Bs[col0 + ka + 1];
            b1.x = Bs[col1 + ka]; b1.y = Bs[col1 + ka + 1];
            // 4 independent accumulation chains: no back-to-back RAW on C
            acc00 = __builtin_amdgcn_wmma_f32_16x16x4_f32(false, a0, false, b0, (short)0, acc00, false, false);
            acc01 = __builtin_amdgcn_wmma_f32_16x16x4_f32(false, a0, false, b1, (short)0, acc01, false, false);
            acc10 = __builtin_amdgcn_wmma_f32_16x16x4_f32(false, a1, false, b0, (short)0, acc10, false, false);
            acc11 = __builtin_amdgcn_wmma_f32_16x16x4_f32(false, a1, false, b1, (short)0, acc11, false, false);
        }
    }
    __syncthreads();

    // Per-lane loss epilogue: 4 subtiles x 8 entries.
    // acc[a][b] entry r maps to (M = wr + a*16 + r + 8*half, N = wc + b*16 + ln)
    float sWP = 0.0f, sWPd = 0.0f, sWN = 0.0f, sWNs = 0.0f;
    #pragma unroll
    for (int ab = 0; ab < 4; ++ab) {
        const int a = ab >> 1, b = ab & 1;
        const v8f acc = (ab == 0) ? acc00 : (ab == 1) ? acc01 : (ab == 2) ? acc10 : acc11;
        const int ncol = wc + b * 16 + ln;
        const int j    = j0 + ncol;
        const float aj  = attJ[ncol];
        const float sqj = sqJ[ncol];
        const int   lj  = labJ[ncol];
        #pragma unroll
        for (int r = 0; r < 8; ++r) {
            const int mrow = wr + a * 16 + r + 8 * half;
            const int i    = i0 + mrow;
            if (i == j) continue;                       // off_diag
            float d2 = sqI[mrow] + sqj - 2.0f * acc[r];
            d2 = fmaxf(d2, EPSV);
            const float dist = sqrtf(d2);
            const float Sn   = fmaxf(ALPHA - dist, EPSV);
            const float Amin = fminf(attI[mrow], aj);
            if (labI[mrow] == lj) {
                const float w = expf(-d2 / SIGMA2) * Amin;   // S * A
                sWP += w; sWPd += w * d2;                    // W_P, W_P*dist^2
            } else {
                const float w = Sn * Amin;                   // S_ * A
                sWN += w; sWNs += w * Sn * Sn;               // W_N, W_N*S_^2
            }
        }
    }

    red[tid] = sWP; red[128 + tid] = sWPd; red[256 + tid] = sWN; red[384 + tid] = sWNs;
    __syncthreads();
    for (int off = 64; off > 0; off >>= 1) {
        if (tid < off) {
            red[tid]       += red[tid + off];
            red[128 + tid] += red[128 + tid + off];
            red[256 + tid] += red[256 + tid + off];
            red[384 + tid] += red[384 + tid + off];
        }
        __syncthreads();
    }
    if (tid == 0) {
        atomicAdd(&accum[0], red[0]);
        atomicAdd(&accum[1], red[128]);
        atomicAdd(&accum[2], red[256]);
        atomicAdd(&accum[3], red[384]);
    }
}

// ---------------- K5: combine ratios into the scalar loss ----------------
__global__ void k_final(const float* __restrict__ accum, float* __restrict__ out) {
    if (threadIdx.x == 0) {
        const float L_P = 0.5f * accum[1] / accum[0];
        const float L_N = 0.5f * accum[3] / accum[2];
        out[0] = (1.0f - L_BAL) * L_P + L_BAL * L_N;
    }
}

extern "C" void kernel_launch(void* const* d_in, const int* in_sizes, int n_in,
                              void* d_out, int out_size, void* d_ws, size_t ws_size,
                              hipStream_t stream) {
    const float*     x      = (const float*)d_in[0];
    const long long* labels = (const long long*)d_in[1];
    const float*     embd   = (const float*)d_in[2];

    const int n = in_sizes[1];                 // 4096
    const int d = in_sizes[0] / n;             // 2048
    const int C = in_sizes[2] / d;             // 1000

    // Workspace layout (floats): xn | enT | sq | atten | accum
    float* ws    = (float*)d_ws;
    float* xn    = ws;
    float* enT   = xn  + (size_t)n * d;
    float* sq    = enT + (size_t)C * d;
    float* atten = sq  + n;
    float* accum = atten + n;

    k_zero_accum<<<1, 32, 0, stream>>>(accum);
    k_rownorm   <<<n, 256, 0, stream>>>(x, xn, sq, d);
    k_colnorm_T <<<C, 256, 0, stream>>>(embd, enT, d, C);
    k_atten     <<<n, 256, 0, stream>>>(xn, enT, labels, atten, d, C);

    dim3 grid(n / BT, n / BT);                 // 64 x 64 blocks
    k_gram_loss <<<grid, 128, 0, stream>>>(xn, sq, atten, labels, accum, n, d);

    k_final     <<<1, 32, 0, stream>>>(accum, (float*)d_out);
}